// VectorQuantizer_72129680769393
// MI455X (gfx1250) — compile-verified
//
#include <hip/hip_runtime.h>

// ---------------------------------------------------------------------------
// VectorQuantizer for MI455X (gfx1250, wave32, WMMA)
//   z:         [16, 64, 32, 32] fp32   (B, C, H, W)
//   embedding: [4096, 64] fp32
// Outputs (concatenated in d_out, fp32):
//   z_q [16,64,32,32] (=1048576), loss (1), indices [16,32,32] (=16384)
// ---------------------------------------------------------------------------

typedef __attribute__((ext_vector_type(2))) float v2f;
typedef __attribute__((ext_vector_type(8))) float v8f;

#define NE       4096      // codebook entries
#define EDIM     64        // vector dim
#define NROWS    16384     // 16*32*32 query vectors
#define HW       1024      // 32*32
#define CSTRIDE  1024      // stride between channels in z (b,c,h,w)
#define BSTRIDE  65536     // 64*1024, stride between batches
#define ZQ_ELEMS 1048576   // 16*64*32*32

// ---------------------------------------------------------------------------
// K1: e2[m] = sum_c embedding[m][c]^2   (tiny: 1 MB read)
// ---------------------------------------------------------------------------
__global__ void vq_e2_kernel(const float* __restrict__ emb,
                             float* __restrict__ e2) {
  int m = blockIdx.x * blockDim.x + threadIdx.x;
  if (m >= NE) return;
  const float4* row = (const float4*)(emb + m * EDIM);
  float s = 0.f;
#pragma unroll
  for (int i = 0; i < EDIM / 4; ++i) {
    float4 v = row[i];
    s += v.x * v.x + v.y * v.y + v.z * v.z + v.w * v.w;
  }
  e2[m] = s;
}

// ---------------------------------------------------------------------------
// K2: fused GEMM + argmin via V_WMMA_F32_16X16X4_F32.
//   s[n][m] = e2[m] - 2 * z[n].e[m]   (z^2 term is row-constant: argmin-safe)
// One wave owns a 16-row A tile (resident VGPRs, pre-scaled by -2) and
// sweeps all 4096 embeddings as 4x 16-wide column tiles per iteration:
// four independent 16-deep accumulation chains -> ILP for the matrix pipe
// and load-latency hiding. Accumulators are seeded with e2[column] so the
// WMMA emits the score directly. Per-lane running (min, idx), then a
// 16-lane shuffle min-reduction with first-index tie-break.
// ---------------------------------------------------------------------------
__global__ void __launch_bounds__(256)
vq_argmin_kernel(const float* __restrict__ z,
                 const float* __restrict__ emb,
                 const float* __restrict__ e2,
                 int* __restrict__ idx_out) {
  const int tid    = threadIdx.x;
  const int lane   = tid & 31;
  const int lane16 = lane & 15;
  const int half   = lane >> 4;          // 0: lanes 0-15, 1: lanes 16-31
  const int wave   = tid >> 5;           // 8 waves per workgroup
  const int n0     = blockIdx.x * 128 + wave * 16;

  // ---- Resident A tile: a[kk] holds A[M=lane16][K=4kk+2*half +{0,1}],
  //      pre-scaled by -2 (so WMMA computes e2 - 2 z.e directly).
  const int row = n0 + lane16;
  const int b   = row >> 10;
  const int hw  = row & (HW - 1);
  const float* zbase = z + b * BSTRIDE + hw;
  v2f a[16];
#pragma unroll
  for (int kk = 0; kk < 16; ++kk) {
    const int c0 = 4 * kk + 2 * half;
    a[kk].x = -2.0f * zbase[c0 * CSTRIDE];
    a[kk].y = -2.0f * zbase[(c0 + 1) * CSTRIDE];
  }

  float best[8];
  int   bidx[8];
#pragma unroll
  for (int v = 0; v < 8; ++v) { best[v] = 3.4e38f; bidx[v] = 0; }

  for (int m0 = 0; m0 < NE; m0 += 64) {
    const int mA = m0 + lane16;          // column owned by this lane, tile A
    const int mB = mA + 16;
    const int mC = mA + 32;
    const int mD = mA + 48;

    v8f cA, cB, cC, cD;
    {
      const float e2A = e2[mA];
      const float e2B = e2[mB];
      const float e2C = e2[mC];
      const float e2D = e2[mD];
#pragma unroll
      for (int v = 0; v < 8; ++v) {
        cA[v] = e2A; cB[v] = e2B; cC[v] = e2C; cD[v] = e2D;
      }
    }

    // B layout: lane L, vgpr v -> B[K = kbase + 2*half + v][N = lane16]
    //          = emb[m][kbase + 2*half + v] : one aligned float2 per step.
    const float* ebA = emb + mA * EDIM + 2 * half;
    const float* ebB = emb + mB * EDIM + 2 * half;
    const float* ebC = emb + mC * EDIM + 2 * half;
    const float* ebD = emb + mD * EDIM + 2 * half;
#pragma unroll
    for (int kk = 0; kk < 16; ++kk) {
      const v2f bA = *(const v2f*)(ebA + 4 * kk);
      const v2f bB = *(const v2f*)(ebB + 4 * kk);
      const v2f bC = *(const v2f*)(ebC + 4 * kk);
      const v2f bD = *(const v2f*)(ebD + 4 * kk);
      cA = __builtin_amdgcn_wmma_f32_16x16x4_f32(false, a[kk], false, bA,
                                                 (short)0, cA, false, false);
      cB = __builtin_amdgcn_wmma_f32_16x16x4_f32(false, a[kk], false, bB,
                                                 (short)0, cB, false, false);
      cC = __builtin_amdgcn_wmma_f32_16x16x4_f32(false, a[kk], false, bC,
                                                 (short)0, cC, false, false);
      cD = __builtin_amdgcn_wmma_f32_16x16x4_f32(false, a[kk], false, bD,
                                                 (short)0, cD, false, false);
    }

    // Running min; ascending-m update order keeps first-index tie-break.
#pragma unroll
    for (int v = 0; v < 8; ++v) {
      if (cA[v] < best[v]) { best[v] = cA[v]; bidx[v] = mA; }
      if (cB[v] < best[v]) { best[v] = cB[v]; bidx[v] = mB; }
      if (cC[v] < best[v]) { best[v] = cC[v]; bidx[v] = mC; }
      if (cD[v] < best[v]) { best[v] = cD[v]; bidx[v] = mD; }
    }
  }

  // ---- Cross-lane min reduction within each 16-lane half (wave32 shuffles).
#pragma unroll
  for (int v = 0; v < 8; ++v) {
#pragma unroll
    for (int off = 1; off < 16; off <<= 1) {
      float ov = __shfl_xor(best[v], off, 32);
      int   oi = __shfl_xor(bidx[v], off, 32);
      if (ov < best[v] || (ov == best[v] && oi < bidx[v])) {
        best[v] = ov; bidx[v] = oi;
      }
    }
  }

  // C/D layout: vgpr v holds row M = v + 8*half -> lanes {0,16} write 8 rows.
  if (lane16 == 0) {
    const int base = n0 + 8 * half;
#pragma unroll
    for (int v = 0; v < 8; ++v) idx_out[base + v] = bidx[v];
  }
}

// ---------------------------------------------------------------------------
// K3: gather z_q = embedding[idx] into (b,c,h,w) layout, emit float indices,
//     and per-block deterministic loss partial sums.
// ---------------------------------------------------------------------------
__global__ void __launch_bounds__(256)
vq_gather_kernel(const float* __restrict__ z,
                 const float* __restrict__ emb,
                 const int* __restrict__ idx,
                 float* __restrict__ zq_out,
                 float* __restrict__ idxf_out,
                 float* __restrict__ partials) {
  const int n  = blockIdx.x * 256 + threadIdx.x;
  const int b  = n >> 10;
  const int hw = n & (HW - 1);
  const int m  = idx[n];
  const float* erow  = emb + m * EDIM;       // gather, L2-resident (1 MB)
  const float* zbase = z + b * BSTRIDE + hw; // coalesced over hw
  float* qbase = zq_out + b * BSTRIDE + hw;  // coalesced over hw
  float acc = 0.f;
#pragma unroll
  for (int c = 0; c < EDIM; ++c) {
    const float e  = erow[c];
    const float zv = zbase[c * CSTRIDE];
    qbase[c * CSTRIDE] = e;
    const float d = zv - e;
    acc += d * d;
  }
  idxf_out[n] = (float)m;

  __shared__ float sred[256];
  sred[threadIdx.x] = acc;
  __syncthreads();
  for (int s = 128; s > 0; s >>= 1) {
    if (threadIdx.x < s) sred[threadIdx.x] += sred[threadIdx.x + s];
    __syncthreads();
  }
  if (threadIdx.x == 0) partials[blockIdx.x] = sred[0];
}

// ---------------------------------------------------------------------------
// K4: fixed-order sum of 64 partials; loss = (1 + 0.25) * mean(diff^2).
// ---------------------------------------------------------------------------
__global__ void vq_loss_kernel(const float* __restrict__ partials,
                               float* __restrict__ loss_out) {
  if (threadIdx.x == 0) {
    float s = 0.f;
    for (int i = 0; i < 64; ++i) s += partials[i];
    loss_out[0] = 1.25f * (s / (float)ZQ_ELEMS);
  }
}

// ---------------------------------------------------------------------------
extern "C" void kernel_launch(void* const* d_in, const int* in_sizes, int n_in,
                              void* d_out, int out_size, void* d_ws,
                              size_t ws_size, hipStream_t stream) {
  const float* z   = (const float*)d_in[0];   // [16,64,32,32]
  const float* emb = (const float*)d_in[1];   // [4096,64]

  float* out  = (float*)d_out;
  float* zq   = out;                 // [0, 1048576)
  float* loss = out + ZQ_ELEMS;      // [1048576]
  float* idxf = out + ZQ_ELEMS + 1;  // [1048577, 1048577+16384)

  char*  ws       = (char*)d_ws;
  float* e2       = (float*)ws;                              // 4096 f32
  int*   idx      = (int*)(ws + NE * sizeof(float));         // 16384 i32
  float* partials = (float*)(ws + NE * sizeof(float)
                                + NROWS * sizeof(int));      // 64 f32

  vq_e2_kernel<<<NE / 256, 256, 0, stream>>>(emb, e2);
  vq_argmin_kernel<<<NROWS / 128, 256, 0, stream>>>(z, emb, e2, idx);
  vq_gather_kernel<<<NROWS / 256, 256, 0, stream>>>(z, emb, idx, zq, idxf,
                                                    partials);
  vq_loss_kernel<<<1, 32, 0, stream>>>(partials, loss);
}